// UnfoldindAndAttention_79164837200037
// MI455X (gfx1250) — compile-verified
//
#include <hip/hip_runtime.h>

// ---------------------------------------------------------------------------
// APPNP-style propagation on MI455X (gfx1250).
// L2-resident gather + non-returning f32 atomics; TDM stages edge-index tiles
// into LDS (async tensor path, TENSORcnt); indices read back via ds_load_b32.
// ---------------------------------------------------------------------------

#define FEAT 64          // D
#define TILE 1024        // edges staged per block
#define BLK  256         // 8 waves of 32

typedef unsigned int u32;
typedef unsigned long long u64;
typedef __attribute__((ext_vector_type(4))) u32 v4u;
typedef __attribute__((ext_vector_type(8))) int v8i;
typedef __attribute__((ext_vector_type(4))) int v4i;

__device__ __forceinline__ u32 lds_byte_off(const void* p) {
  // Flat shared pointer: low 32 bits are the workgroup-relative LDS byte addr.
  return (u32)(u64)(uintptr_t)p;
}

// 1-D TDM copy: tile_elems 4-byte elements from global gptr to the LDS byte address lds_off.
// tensor_dim0 = elems_remaining so the tail reads OOB-zero instead of faulting.
__device__ __forceinline__ void tdm_load_1d(u32 lds_off, const void* gptr,
                                            u32 elems_remaining, u32 tile_elems) {
  u64 ga = (u64)(uintptr_t)gptr;
  v4u g0;
  g0.x = 1u;                                             // count=1, user D#
  g0.y = lds_off;                                        // lds_addr (bytes)
  g0.z = (u32)ga;                                        // global_addr[31:0]
  g0.w = (u32)((ga >> 32) & 0x01FFFFFFull) | (2u << 30); // global_addr[56:32] | type=2
  v8i g1;
  g1[0] = (int)(2u << 16);                               // data_size = 4B; no multicast
  g1[1] = (int)((elems_remaining & 0xFFFFu) << 16);      // tensor_dim0[15:0] @ [63:48]
  g1[2] = (int)(((elems_remaining >> 16) & 0xFFFFu)      // tensor_dim0[31:16] @ [79:64]
                | (1u << 16));                           // tensor_dim1 = 1
  g1[3] = (int)((tile_elems & 0xFFFFu) << 16);           // tile_dim0 @ [127:112]
  g1[4] = 0;                                             // tile_dim1=0, tile_dim2=0 (1-D)
  g1[5] = (int)elems_remaining;                          // tensor_dim0_stride (unused, 1-D)
  g1[6] = 0;
  g1[7] = 0;
  v4i z4 = {0, 0, 0, 0};
#if defined(__clang_major__) && (__clang_major__ >= 23)
  v8i z8 = {0, 0, 0, 0, 0, 0, 0, 0};
  __builtin_amdgcn_tensor_load_to_lds(g0, g1, z4, z4, z8, 0);
#else
  __builtin_amdgcn_tensor_load_to_lds(g0, g1, z4, z4, 0);
#endif
}

__device__ __forceinline__ void atomic_add_f32(float* p, float v) {
  unsafeAtomicAdd(p, v);   // hardware global_atomic_add_f32, non-returning
}

// ---------------------------------------------------------------------------
// Kernels
// ---------------------------------------------------------------------------

__global__ __launch_bounds__(BLK) void zero_f4_kernel(float4* __restrict__ p, int n4) {
  int i = blockIdx.x * BLK + threadIdx.x;
  if (i < n4) p[i] = make_float4(0.f, 0.f, 0.f, 0.f);
}

__global__ __launch_bounds__(BLK) void deg_kernel(const int* __restrict__ dst,
                                                  float* __restrict__ deg, int E) {
  int e = blockIdx.x * BLK + threadIdx.x;
  if (e < E) atomic_add_f32(&deg[dst[e]], 1.0f);
}

// inv0/inv1 alias deg0/deg1 storage: read degree, overwrite with 1/sqrt(deg).
__global__ __launch_bounds__(BLK) void coef_kernel(float* __restrict__ inv0,
                                                   float* __restrict__ inv1,
                                                   float* __restrict__ w, int N) {
  int i = blockIdx.x * BLK + threadIdx.x;
  if (i >= N) return;
  float d0 = inv0[i];
  float d1 = inv1[i];
  inv0[i] = d0 > 0.f ? 1.0f / sqrtf(d0) : 0.f;
  inv1[i] = d1 > 0.f ? 1.0f / sqrtf(d1) : 0.f;
  w[i] = 1.0f / (d0 + d1);   // LAM0 = 1
}

// Fused dual-graph scatter: blocks [0, gTile) handle graph 0, [gTile, 2*gTile)
// handle graph 1.  agg[dst] += Y[src] * inv[src]; 16 lanes/edge, float4/lane.
__global__ __launch_bounds__(BLK) void scatter2_kernel(const float* __restrict__ Y,
                                                       const int* __restrict__ src0,
                                                       const int* __restrict__ dst0,
                                                       const float* __restrict__ inv0,
                                                       float* __restrict__ agg0,
                                                       const int* __restrict__ src1,
                                                       const int* __restrict__ dst1,
                                                       const float* __restrict__ inv1,
                                                       float* __restrict__ agg1,
                                                       int E, int gTile) {
  // Single allocation: sIdx[0..TILE) = src indices, sIdx[TILE..2*TILE) = dst
  // indices, so dst sits exactly TILE*4 bytes past src (DS offset immediate).
  __shared__ int sIdx[2 * TILE];

  const int g = (blockIdx.x >= gTile) ? 1 : 0;
  const int tileIdx = blockIdx.x - g * gTile;
  const int* __restrict__ src = g ? src1 : src0;
  const int* __restrict__ dst = g ? dst1 : dst0;
  const float* __restrict__ invs = g ? inv1 : inv0;
  float* __restrict__ agg = g ? agg1 : agg0;

  const int e0 = tileIdx * TILE;
  const int rem = E - e0;
  const u32 sBase = lds_byte_off(sIdx);

  // Stage this block's index tile into LDS: each of the 8 waves issues two
  // 1-D TDM copies (128 src + 128 dst indices) and waits its own TENSORcnt.
  {
    const u32 wave = (u32)(threadIdx.x >> 5);      // wave-uniform
    const u32 slice = TILE / 8;                    // 128 edges per wave
    const u32 off = wave * slice;
    const u32 wrem = (rem > (int)off) ? (u32)(rem - (int)off) : 0u;
    tdm_load_1d(sBase + off * 4u, src + e0 + off, wrem, slice);
    tdm_load_1d(sBase + (u32)TILE * 4u + off * 4u, dst + e0 + off, wrem, slice);
    __builtin_amdgcn_s_wait_tensorcnt(0);
  }
  // Compiler barrier: TDM modified LDS behind the compiler's back.
  asm volatile("" ::: "memory");
  __syncthreads();

  const int valid = rem < TILE ? rem : TILE;
  const int lane16 = threadIdx.x & 15;             // which float4 of the 64-f row
  const float4* __restrict__ Y4 = (const float4*)Y;

  for (int t = (threadIdx.x >> 4); t < valid; t += (BLK / 16)) {
    const u32 a = sBase + (u32)t * 4u;
    int s, d;
    // Deterministic DS path: both index loads through one address register,
    // dst via the 16-bit DS offset immediate.  Earlyclobber outputs: the
    // first load's writeback must not alias the address reg of the second.
    asm volatile("ds_load_b32 %0, %2\n\t"
                 "ds_load_b32 %1, %2 offset:4096\n\t"
                 "s_wait_dscnt 0x0"
                 : "=&v"(s), "=&v"(d)
                 : "v"(a));
    const float sc = invs[s];
    float4 v = Y4[(size_t)s * (FEAT / 4) + lane16];
    float* out = agg + (size_t)d * FEAT + lane16 * 4;
    atomic_add_f32(out + 0, v.x * sc);
    atomic_add_f32(out + 1, v.y * sc);
    atomic_add_f32(out + 2, v.z * sc);
    atomic_add_f32(out + 3, v.w * sc);
  }
}

// Y' = Y - ALP*( w*(Y-x) + LAM*(Y - inv0*agg0) - (LAM_K/K)*(Y - inv1*agg1) )
// ALP=0.5, LAM=1, LAM_K/K=0.125.  Also re-zeroes agg0/agg1 for the next step.
__global__ __launch_bounds__(BLK) void update_kernel(const float* __restrict__ Ycur,
                                                     const float* __restrict__ X,
                                                     float* __restrict__ agg0,
                                                     float* __restrict__ agg1,
                                                     const float* __restrict__ inv0,
                                                     const float* __restrict__ inv1,
                                                     const float* __restrict__ w,
                                                     float* __restrict__ Ynext, int N) {
  int i = blockIdx.x * BLK + threadIdx.x;          // float4 index, N*16 total
  if (i >= N * (FEAT / 4)) return;
  const int n = i >> 4;

  float4 y  = ((const float4*)Ycur)[i];
  float4 xx = ((const float4*)X)[i];
  float4 a0 = ((float4*)agg0)[i];
  float4 a1 = ((float4*)agg1)[i];
  const float i0 = inv0[n], i1 = inv1[n], wn = w[n];

  float4 r;
  r.x = y.x - 0.5f * (wn * (y.x - xx.x) + (y.x - i0 * a0.x) - 0.125f * (y.x - i1 * a1.x));
  r.y = y.y - 0.5f * (wn * (y.y - xx.y) + (y.y - i0 * a0.y) - 0.125f * (y.y - i1 * a1.y));
  r.z = y.z - 0.5f * (wn * (y.z - xx.z) + (y.z - i0 * a0.z) - 0.125f * (y.z - i1 * a1.z));
  r.w = y.w - 0.5f * (wn * (y.w - xx.w) + (y.w - i0 * a0.w) - 0.125f * (y.w - i1 * a1.w));

  ((float4*)Ynext)[i] = r;
  const float4 z = make_float4(0.f, 0.f, 0.f, 0.f);
  ((float4*)agg0)[i] = z;                          // ready for next step's scatter
  ((float4*)agg1)[i] = z;
}

// ---------------------------------------------------------------------------
// Launcher
// ---------------------------------------------------------------------------

extern "C" void kernel_launch(void* const* d_in, const int* in_sizes, int n_in,
                              void* d_out, int out_size, void* d_ws, size_t ws_size,
                              hipStream_t stream) {
  const float* x   = (const float*)d_in[0];
  const int* src0  = (const int*)d_in[1];
  const int* dst0  = (const int*)d_in[2];
  const int* src1  = (const int*)d_in[3];
  const int* dst1  = (const int*)d_in[4];

  const int ND = in_sizes[0];        // N * 64
  const int N  = ND / FEAT;
  const int E  = in_sizes[1];
  float* out = (float*)d_out;

  // Workspace carve-out (256B aligned slices).
  char* ws = (char*)d_ws;
  auto carve = [&](size_t bytes) -> float* {
    float* p = (float*)ws;
    ws += (bytes + 255) & ~(size_t)255;
    return p;
  };
  float* inv0 = carve((size_t)N * 4);     // deg0 then 1/sqrt(deg0)
  float* inv1 = carve((size_t)N * 4);     // deg1 then 1/sqrt(deg1)
  float* w    = carve((size_t)N * 4);     // LAM0/(deg0+deg1)
  float* agg0 = carve((size_t)ND * 4);
  float* agg1 = carve((size_t)ND * 4);
  float* Ya   = carve((size_t)ND * 4);
  float* Yb   = carve((size_t)ND * 4);

  const int n4N  = N / 4;                 // N divisible by 4
  const int n4ND = ND / 4;
  const int gN4   = (n4N + BLK - 1) / BLK;
  const int gND4  = (n4ND + BLK - 1) / BLK;
  const int gE    = (E + BLK - 1) / BLK;
  const int gN    = (N + BLK - 1) / BLK;
  const int gTile = (E + TILE - 1) / TILE;
  const int gUpd  = (N * (FEAT / 4) + BLK - 1) / BLK;

  // Degrees + coefficients (once per call).
  zero_f4_kernel<<<gN4, BLK, 0, stream>>>((float4*)inv0, n4N);
  zero_f4_kernel<<<gN4, BLK, 0, stream>>>((float4*)inv1, n4N);
  deg_kernel<<<gE, BLK, 0, stream>>>(dst0, inv0, E);
  deg_kernel<<<gE, BLK, 0, stream>>>(dst1, inv1, E);
  coef_kernel<<<gN, BLK, 0, stream>>>(inv0, inv1, w, N);

  // Accumulators start at zero; update_kernel re-zeroes them each step.
  zero_f4_kernel<<<gND4, BLK, 0, stream>>>((float4*)agg0, n4ND);
  zero_f4_kernel<<<gND4, BLK, 0, stream>>>((float4*)agg1, n4ND);

  for (int s = 0; s < 8; ++s) {
    const float* Yin = (s == 0) ? x : ((((s - 1) & 1) == 0) ? Ya : Yb);
    float* Yout = (s == 7) ? out : (((s & 1) == 0) ? Ya : Yb);
    scatter2_kernel<<<2 * gTile, BLK, 0, stream>>>(Yin, src0, dst0, inv0, agg0,
                                                   src1, dst1, inv1, agg1, E, gTile);
    update_kernel<<<gUpd, BLK, 0, stream>>>(Yin, x, agg0, agg1, inv0, inv1, w, Yout, N);
  }
}